// QuantumPerceptron_67018669687158
// MI455X (gfx1250) — compile-verified
//
#include <hip/hip_runtime.h>
#include <math.h>

typedef float v2f __attribute__((ext_vector_type(2)));
typedef float v4f __attribute__((ext_vector_type(4)));
typedef float v8f __attribute__((ext_vector_type(8)));

// ======================================================================
// 1) Quantum front-end, collapsed analytically:
//    r[t] = C * sum_m state[m]^2 * M_m(t)^4,  r[9] = 1
//    C = prod over all 40 b-params of cos^2(b)
//    M_m(t) = cos^2(t*w) + (d^2/w^2) sin^2(t*w), d=4-2*popc(m), w=sqrt(d^2+100)
//    Double precision to match jax_enable_x64, cast to f32.
// ======================================================================
__global__ void quantum_r_kernel(const float* __restrict__ state,
                                 const float* __restrict__ params,
                                 float* __restrict__ r_out) {
  const int t = threadIdx.x;
  if (t >= 10) return;
  if (t == 9) { r_out[9] = 1.0f; return; }
  const float* b = params + 40;              // params[1] == b, shape (4,2,5)
  double C = 1.0;
  for (int i = 0; i < 40; ++i) {
    double c = cos((double)b[i]);
    C *= c * c;
  }
  const double tt = 0.01 * (double)(t + 1);  // times 0.01..0.09
  double acc = 0.0;
  for (int m = 0; m < 16; ++m) {
    const int p = __popc(m);
    const double d  = 4.0 - 2.0 * (double)p;
    const double w2 = d * d + 100.0;
    const double w  = sqrt(w2);
    const double cw = cos(tt * w), sw = sin(tt * w);
    const double M  = cw * cw + (d * d / w2) * sw * sw;
    const double M2 = M * M;
    const double s  = (double)state[m];
    acc += s * s * M2 * M2;
  }
  r_out[t] = (float)(C * acc);
}

// ======================================================================
// 2) Tiny first layer: h1 = gelu(W1 @ r + b1), W1 is (8192,10) row-major.
// ======================================================================
__device__ __forceinline__ float gelu_exact(float x) {
  return 0.5f * x * (1.0f + erff(x * 0.70710678118654752f));
}

__global__ void layer1_kernel(const float* __restrict__ W1,
                              const float* __restrict__ b1,
                              const float* __restrict__ r,
                              float* __restrict__ h) {
  const int j = blockIdx.x * blockDim.x + threadIdx.x;
  if (j >= 8192) return;
  float s = b1[j];
#pragma unroll
  for (int k = 0; k < 10; ++k) s += r[k] * W1[j * 10 + k];
  h[j] = gelu_exact(s);
}

// ======================================================================
// 3) HBM-bound f32 GEMV via V_WMMA_F32_16X16X4_F32, software-pipelined.
//    out[j] = act( sum_k x[k]*W[j,k] + bias[j] ),  W row-major (N,K).
//    Block = 256 threads (8 waves), one 16-row tile per block, K split
//    8 ways across waves (KW = K/8). Per 64-wide chunk:
//      prefetch regs (next chunk, nontemporal) overlaps 16x v_wmma of the
//      current chunk; LDS ops are same-wave in-order so store->load through
//      LDS needs no barrier.
//    A layout (ISA 7.12.2, 16x4 f32): VGPR0 = K=0 (lanes 0-15)/K=2 (16-31),
//    VGPR1 = K=1/K=3 -> per-lane b64 read at column 4w+2*hi.
//    B (4x16): same K split; all 16 N columns carry broadcast x.
//    D: VGPR v = row v (lanes 0-15) / row v+8 (lanes 16-31).
// ======================================================================
#define KC   64                   // K chunk per pipeline stage
#define TS   68                   // LDS row stride (floats): bank=(4m+c)%64 conflict-free, 16B rows
#define NW   8                    // waves per block (K split)
#define WREG (16 * TS + KC)       // per-wave LDS floats = 1152

__global__ __launch_bounds__(256) void wmma_gemv_kernel(
    const float* __restrict__ W, const float* __restrict__ bias,
    const float* __restrict__ x, float* __restrict__ out,
    int N, int K, int act /*0=gelu, 1=tanh*/) {
  __shared__ float lds[NW * WREG + 16 * NW];
  float* red = lds + NW * WREG;

  const int tid  = threadIdx.x;
  const int wave = tid >> 5;
  const int lane = tid & 31;
  const int m    = lane & 15;
  const int hi   = lane >> 4;            // half-wave 0/1
  const int row0 = blockIdx.x * 16;

  float* wl = lds + wave * WREG;         // this wave's W tile (16 x KC)
  float* xl = wl + 16 * TS;              // this wave's x chunk (KC)

  const int KW    = K / NW;              // per-wave K span (1024 for K=8192)
  const int kbase = wave * KW;
  const int nch   = KW / KC;             // chunks per wave (16)

  v8f acc = {0.f, 0.f, 0.f, 0.f, 0.f, 0.f, 0.f, 0.f};
  v4f wreg[8];                           // prefetch buffer: 8 loads x 2 rows
  v2f xreg;

  // ---- prefetch chunk 0 ----
  {
    const int kg = kbase;
    xreg = *(const v2f*)(x + kg + lane * 2);
#pragma unroll
    for (int rr = 0; rr < 8; ++rr) {
      const int row = row0 + rr * 2 + hi;
      v4f wv = {0.f, 0.f, 0.f, 0.f};
      if (row < N)
        wv = __builtin_nontemporal_load(
            (const v4f*)(W + (size_t)row * (size_t)K + (size_t)(kg + m * 4)));
      wreg[rr] = wv;
    }
  }

  for (int c = 0; c < nch; ++c) {
    // ---- commit prefetched chunk to LDS ----
    *(v2f*)(xl + lane * 2) = xreg;
#pragma unroll
    for (int rr = 0; rr < 8; ++rr)
      *(v4f*)(wl + (rr * 2 + hi) * TS + m * 4) = wreg[rr];

    // ---- issue prefetch for next chunk (overlaps compute below) ----
    if (c + 1 < nch) {
      const int kg = kbase + (c + 1) * KC;
      xreg = *(const v2f*)(x + kg + lane * 2);
#pragma unroll
      for (int rr = 0; rr < 8; ++rr) {
        const int row = row0 + rr * 2 + hi;
        v4f wv = {0.f, 0.f, 0.f, 0.f};
        if (row < N)
          wv = __builtin_nontemporal_load(
              (const v4f*)(W + (size_t)row * (size_t)K + (size_t)(kg + m * 4)));
        wreg[rr] = wv;
      }
    }

    // ---- 16x WMMA over current chunk (same-wave LDS is in-order) ----
#pragma unroll
    for (int w = 0; w < 16; ++w) {
      const int kk = 4 * w + 2 * hi;               // this lane's K pair
      v2f a  = *(const v2f*)(wl + m * TS + kk);    // W[row0+m, kg+kk .. +1]
      v2f bb = *(const v2f*)(xl + kk);             // x[kg+kk .. +1] (broadcast)
      acc = __builtin_amdgcn_wmma_f32_16x16x4_f32(
          /*neg_a=*/false, a, /*neg_b=*/false, bb,
          /*c_mod=*/(short)0, acc, /*reuse_a=*/false, /*reuse_b=*/false);
    }
  }

  // lanes 0 and 16 of each wave hold the 16 per-wave partial row sums
  if (m == 0) {
#pragma unroll
    for (int v = 0; v < 8; ++v)
      red[wave * 16 + hi * 8 + v] = acc[v];
  }
  __syncthreads();

  if (tid < 16) {
    const int row = row0 + tid;
    if (row < N) {
      float s = bias[row];
#pragma unroll
      for (int wv = 0; wv < NW; ++wv) s += red[wv * 16 + tid];
      out[row] = (act == 0) ? gelu_exact(s) : tanhf(s);
    }
  }
}

// ======================================================================
// Launch: quantum -> layer1 -> Wm0 -> Wm1 -> W2(tanh) -> d_out (1000 f32)
// ======================================================================
extern "C" void kernel_launch(void* const* d_in, const int* in_sizes, int n_in,
                              void* d_out, int out_size, void* d_ws, size_t ws_size,
                              hipStream_t stream) {
  const float* state  = (const float*)d_in[0];
  const float* params = (const float*)d_in[1];
  const float* W1  = (const float*)d_in[2];
  const float* b1  = (const float*)d_in[3];
  const float* Wm0 = (const float*)d_in[4];
  const float* bm0 = (const float*)d_in[5];
  const float* Wm1 = (const float*)d_in[6];
  const float* bm1 = (const float*)d_in[7];
  const float* W2  = (const float*)d_in[8];
  const float* b2  = (const float*)d_in[9];

  float* ws = (float*)d_ws;
  float* r  = ws;                  // 16 floats reserved (10 used)
  float* h1 = ws + 16;             // 8192
  float* h2 = ws + 16 + 8192;      // 8192
  float* h3 = ws + 16 + 2 * 8192;  // 8192
  float* outp = (float*)d_out;     // 1000 f32

  quantum_r_kernel<<<1, 32, 0, stream>>>(state, params, r);
  layer1_kernel<<<32, 256, 0, stream>>>(W1, b1, r, h1);
  wmma_gemv_kernel<<<512, 256, 0, stream>>>(Wm0, bm0, h1, h2, 8192, 8192, 0);
  wmma_gemv_kernel<<<512, 256, 0, stream>>>(Wm1, bm1, h2, h3, 8192, 8192, 0);
  wmma_gemv_kernel<<<63, 256, 0, stream>>>(W2, b2, h3, outp, 1000, 8192, 1);
}